// RelaxedTripletMarginLoss_12395275616912
// MI455X (gfx1250) — compile-verified
//
#include <hip/hip_runtime.h>
#include <math.h>

#define NN 384
#define DD 512
#define KK 4
#define MARGIN   0.2f
#define MARGIN2  0.2f
#define INV_TAU  10.0f

typedef __attribute__((ext_vector_type(2))) float v2f;
typedef __attribute__((ext_vector_type(8))) float v8f;

// ---------------------------------------------------------------------------
// Kernel 1: 16x16 distance tiles via V_WMMA_F32_16X16X4_F32.
// grid = (24, 24, 2); z==0 -> source avg distances (loop k=0..3), z==1 -> student.
// One wave32 per tile; EXEC is all-ones (no divergence around the WMMA).
// ---------------------------------------------------------------------------
__global__ __launch_bounds__(32)
void dist_tiles_kernel(const float* __restrict__ src, const float* __restrict__ emb,
                       float* __restrict__ sd, float* __restrict__ dd)
{
    const int row0 = blockIdx.x * 16;
    const int col0 = blockIdx.y * 16;
    const int lane = threadIdx.x;     // 0..31
    const int lsub = lane & 15;
    const int hi   = lane >> 4;       // 0: lanes 0-15, 1: lanes 16-31

    __shared__ float rowsq[16];
    __shared__ float colsq[16];

    float avg[8];
#pragma unroll
    for (int r = 0; r < 8; ++r) avg[r] = 0.0f;

    const int   nmat  = (blockIdx.z == 0) ? KK : 1;
    const float scale = (blockIdx.z == 0) ? (1.0f / (float)KK) : 1.0f;

    for (int k = 0; k < nmat; ++k) {
        const float* X = (blockIdx.z == 0) ? (src + (size_t)k * NN * DD) : emb;

        // Squared norms: lanes 0-15 cover the 16 tile rows, lanes 16-31 the 16 cols.
        {
            const float* xr = X + (size_t)((hi == 0 ? row0 : col0) + lsub) * DD;
            float s = 0.0f;
            for (int j = 0; j < DD; j += 2) {
                v2f v = *(const v2f*)(xr + j);
                s = fmaf(v.x, v.x, s);
                s = fmaf(v.y, v.y, s);
            }
            if (hi == 0) rowsq[lsub] = s; else colsq[lsub] = s;
        }
        __syncthreads();

        // Gram tile: acc = A(16xD) * A(16xD)^T restricted to this tile.
        // A frag (16x4 f32): lane<16 holds {K=kk,kk+1} of row row0+lsub,
        //                    lane>=16 holds {K=kk+2,kk+3}.  B frag mirrors with col0.
        v8f acc = {};
        const float* Ar = X + (size_t)(row0 + lsub) * DD;
        const float* Br = X + (size_t)(col0 + lsub) * DD;
        for (int kk = 0; kk < DD; kk += 4) {
            v2f a = *(const v2f*)(Ar + kk + 2 * hi);
            v2f b = *(const v2f*)(Br + kk + 2 * hi);
            acc = __builtin_amdgcn_wmma_f32_16x16x4_f32(
                /*neg_a=*/false, a, /*neg_b=*/false, b,
                /*c_mod=*/(short)0, acc, /*reuse_a=*/false, /*reuse_b=*/false);
        }

        // d = sqrt(max(|x|^2 + |y|^2 - 2*gram, 0)); C/D layout: M = r + 8*hi, N = lsub.
#pragma unroll
        for (int r = 0; r < 8; ++r) {
            const int   M  = r + 8 * hi;
            const float d2 = rowsq[M] + colsq[lsub] - 2.0f * acc[r];
            avg[r] += scale * sqrtf(fmaxf(d2, 0.0f));
        }
        __syncthreads();
    }

    float* out = (blockIdx.z == 0) ? sd : dd;
#pragma unroll
    for (int r = 0; r < 8; ++r) {
        const int M = r + 8 * hi;
        out[(size_t)(row0 + M) * NN + (col0 + lsub)] = avg[r];
    }
}

// ---------------------------------------------------------------------------
// Kernel 2: O(N^3) triplet reduction. One block per anchor a; rows cached in LDS.
// Hot loop is VALU + v_exp_f32 only — zero global traffic.
// ---------------------------------------------------------------------------
__global__ __launch_bounds__(256)
void triplet_kernel(const float* __restrict__ sd, const float* __restrict__ dd,
                    float* __restrict__ partial)
{
    const int a   = blockIdx.x;
    const int tid = threadIdx.x;

    __shared__ float sdrow[NN];
    __shared__ float drow[NN];
    for (int i = tid; i < NN; i += 256) {
        sdrow[i] = sd[(size_t)a * NN + i];
        drow[i]  = dd[(size_t)a * NN + i];
    }
    __syncthreads();

    float num = 0.0f, den = 0.0f;
    for (int p = 0; p < NN; ++p) {
        const float sd_ap = sdrow[p];
        const float d_ap  = drow[p];
        for (int n = tid; n < NN; n += 256) {
            const float x  = (sdrow[n] - sd_ap - MARGIN2) * INV_TAU;
            const float w  = 1.0f / (1.0f + __expf(-x));          // sigmoid
            const float tl = fmaxf(MARGIN - (drow[n] - d_ap), 0.0f); // hinge
            num = fmaf(tl, w, num);
            den += w;
        }
    }

    __shared__ float rn[256];
    __shared__ float rd[256];
    rn[tid] = num; rd[tid] = den;
    __syncthreads();
    for (int s = 128; s > 0; s >>= 1) {
        if (tid < s) { rn[tid] += rn[tid + s]; rd[tid] += rd[tid + s]; }
        __syncthreads();
    }
    if (tid == 0) { partial[2 * a] = rn[0]; partial[2 * a + 1] = rd[0]; }
}

// ---------------------------------------------------------------------------
// Kernel 3: deterministic final reduce + safe division.
// ---------------------------------------------------------------------------
__global__ __launch_bounds__(256)
void finalize_kernel(const float* __restrict__ partial, float* __restrict__ out)
{
    const int tid = threadIdx.x;
    float num = 0.0f, den = 0.0f;
    for (int i = tid; i < NN; i += 256) {
        num += partial[2 * i];
        den += partial[2 * i + 1];
    }
    __shared__ float rn[256];
    __shared__ float rd[256];
    rn[tid] = num; rd[tid] = den;
    __syncthreads();
    for (int s = 128; s > 0; s >>= 1) {
        if (tid < s) { rn[tid] += rn[tid + s]; rd[tid] += rd[tid + s]; }
        __syncthreads();
    }
    if (tid == 0) {
        const float d = rd[0];
        out[0] = (d != 0.0f) ? (rn[0] / d) : 0.0f;
    }
}

// ---------------------------------------------------------------------------
extern "C" void kernel_launch(void* const* d_in, const int* in_sizes, int n_in,
                              void* d_out, int out_size, void* d_ws, size_t ws_size,
                              hipStream_t stream)
{
    (void)in_sizes; (void)n_in; (void)out_size; (void)ws_size;

    const float* src = (const float*)d_in[0];   // [K, N, D] f32
    const float* emb = (const float*)d_in[1];   // [N, D]    f32

    float* sd      = (float*)d_ws;              // [N, N] source avg distances
    float* dd      = sd + (size_t)NN * NN;      // [N, N] student distances
    float* partial = dd + (size_t)NN * NN;      // [N, 2] per-anchor partials
    float* out     = (float*)d_out;             // scalar loss

    dist_tiles_kernel<<<dim3(NN / 16, NN / 16, 2), 32, 0, stream>>>(src, emb, sd, dd);
    triplet_kernel<<<NN, 256, 0, stream>>>(sd, dd, partial);
    finalize_kernel<<<1, 256, 0, stream>>>(partial, out);
}